// metamorphoses_76544907149719
// MI455X (gfx1250) — compile-verified
//
#include <hip/hip_runtime.h>
#include <hip/hip_bf16.h>
#include <cstdint>

// ---------------------------------------------------------------------------
// Metamorphosis shooting (L=10) on 96^3 fp32 volume, MI455X (gfx1250, wave32).
// - Gaussian RKHS smoothing = banded-Toeplitz tile matmul on the fp32 WMMA
//   pipe (V_WMMA_F32_16X16X4_F32); B-operand tiles staged Global->LDS by the
//   Tensor Data Mover (tensor_load_to_lds + s_wait_tensorcnt); A-fragments
//   preloaded from a per-lane table (12 coalesced b64 loads, zero VALU setup).
// - Everything else: fused stencil/gather kernels. Bandwidth-bound
//   (~2 GB HBM traffic/launch -> ~100us floor at 23.3 TB/s).
// ---------------------------------------------------------------------------

#define DIMV 96
#define NVOX (96 * 96 * 96)        // 884736
#define S1   96
#define S2   (96 * 96)             // 9216
#define LSTEPS 10

typedef __attribute__((ext_vector_type(2))) float v2f;
typedef __attribute__((ext_vector_type(8))) float v8f;
typedef unsigned int u32x4 __attribute__((ext_vector_type(4)));
typedef int i32x4 __attribute__((ext_vector_type(4)));
typedef int i32x8 __attribute__((ext_vector_type(8)));

#ifndef __has_builtin
#define __has_builtin(x) 0
#endif
#if defined(__AMDGCN__) && __has_builtin(__builtin_amdgcn_tensor_load_to_lds) && \
    __has_builtin(__builtin_amdgcn_s_wait_tensorcnt)
#define HAVE_TDM 1
#else
#define HAVE_TDM 0
#endif

// Normalized Gaussian taps, sigma=3, radius 9 (k[t] = exp(-0.5(t/3)^2)/sum)
__device__ __constant__ float GK19[19] = {
    0.0014794f, 0.0038043f, 0.0087534f, 0.0180225f, 0.0332077f,
    0.0547503f, 0.0807754f, 0.1066377f, 0.1259798f, 0.1331760f,
    0.1259798f, 0.1066377f, 0.0807754f, 0.0547503f, 0.0332077f,
    0.0180225f, 0.0087534f, 0.0038043f, 0.0014794f};

__device__ __forceinline__ float gk(int t) {
    return (t >= -9 && t <= 9) ? GK19[t + 9] : 0.0f;
}

// ---------------------------------------------------------------------------
// A-fragment table: 3 Toeplitz offsets x 4 K-chunks, per-lane v2f layout so a
// blur wave loads fragment (o,q) as one coalesced b64:
//   AW[((o*4 + q) * 32 + lane) * 2 + {0,1}]
// A layout (16x4 fp32 WMMA): lanes 0-15 rows 0-15 hold K=0,1; lanes 16-31 K=2,3.
// ---------------------------------------------------------------------------
__global__ __launch_bounds__(384) void ainit_kernel(float* __restrict__ aw) {
    int i = threadIdx.x;          // 384 threads: [o:3][q:4][lane:32]
    int lane = i & 31;
    int q = (i >> 5) & 3;
    int o = i >> 7;
    int row = lane & 15;
    int kb = (lane >> 4) << 1;
    int c0 = 4 * q + kb;
    aw[(size_t)i * 2 + 0] = gk(16 * (o - 1) + c0 - row);
    aw[(size_t)i * 2 + 1] = gk(16 * (o - 1) + c0 + 1 - row);
}

// One band-tile MAC: 4 chained 16x16x4 fp32 WMMAs (K=16 slab of the Toeplitz
// product). Compile-time A fragment selection -> direct register operands.
#if HAVE_TDM
template <bool CONTIG>
__device__ __forceinline__ v8f tile_mac(v8f acc, const v2f (&Ao)[4],
                                        const float* lbase, int nAxis,
                                        int aloc, int kb, int col) {
#pragma unroll
    for (int q = 0; q < 4; ++q) {
        const int kk = 4 * q + kb;
        v2f b;
        if (CONTIG) {
            int i0 = col * nAxis + aloc + kk;
            b.x = lbase[i0];
            b.y = lbase[i0 + 1];
        } else {
            int i0 = (aloc + kk) * 16 + col;
            b.x = lbase[i0];
            b.y = lbase[i0 + 16];
        }
        acc = __builtin_amdgcn_wmma_f32_16x16x4_f32(
            false, Ao[q], false, b, (short)0, acc, false, false);
    }
    return acc;
}
#else
template <int AS, int CS>
__device__ __forceinline__ v8f tile_mac(v8f acc, const v2f (&Ao)[4],
                                        const float* gbase,
                                        int aloc, int kb, int col) {
#pragma unroll
    for (int q = 0; q < 4; ++q) {
        const int kk = 4 * q + kb;
        const float* bp = gbase + (size_t)(aloc + kk) * AS + (size_t)col * CS;
        v2f b;
        b.x = bp[0];
        b.y = bp[AS];
        acc = __builtin_amdgcn_wmma_f32_16x16x4_f32(
            false, Ao[q], false, b, (short)0, acc, false, false);
    }
    return acc;
}
#endif

// ---------------------------------------------------------------------------
// Separable Gaussian blur along one axis via fp32 WMMA.
// out = M @ X per 96-long line; Toeplitz band (halfwidth 9) -> output tile i
// needs input tiles j in {i-1,i,i+1} (12 static WMMAs, edge tiles 8).
// One wave per (channel, plane, colTile, axisTile); per-wave input footprint
// is a 2D tile (contig extent x lines @ line stride) DMA'd to LDS by the TDM.
// ---------------------------------------------------------------------------
template <int AS, int CS, int PS, bool CONTIG>
__global__ __launch_bounds__(128) void blur_axis_wmma(
    const float* __restrict__ in, float* __restrict__ out,
    const float* __restrict__ aw) {
#if HAVE_TDM
    __shared__ float tile[4][48 * 16];   // 12 KB: one 48x16 fp32 tile per wave
#endif
    const int lane = threadIdx.x & 31;
    // wave coords forced scalar: SALU decomposition + s_cbranch tile guards
    const int wave = __builtin_amdgcn_readfirstlane((int)(threadIdx.x >> 5));
    const int waveId = (int)blockIdx.x * 4 + wave;

    const int axisTile = waveId % 6;
    int t = waveId / 6;
    const int colTile = t % 6;
    t /= 6;
    const int plane = t % DIMV;
    const int ch = t / DIMV;
    if (ch >= 3) return;

    const int jStart = max(axisTile - 1, 0);
    const int jEnd   = min(axisTile + 1, 5);
    const int nAxis  = (jEnd - jStart + 1) * 16;   // 32 or 48

    const size_t elemOff = (size_t)ch * NVOX + (size_t)plane * PS
                         + (size_t)colTile * 16 * CS
                         + (size_t)jStart * 16 * AS;

#if HAVE_TDM
    {   // Issue the tensor DMA first; its latency hides the A-table loads.
        const uint64_t gaddr = (uint64_t)(const void*)(in + elemOff);
        const uint32_t ldsOff = (uint32_t)(uintptr_t)&tile[wave][0];
        const uint32_t tdim0 = CONTIG ? (uint32_t)nAxis : 16u;  // contig extent
        const uint32_t nLine = CONTIG ? 16u : (uint32_t)nAxis;  // #lines
        const uint32_t lstr  = (uint32_t)(CONTIG ? CS : AS);    // line stride

        u32x4 g0;
        g0[0] = 1u;                                   // count=1 (user descriptor)
        g0[1] = ldsOff;                               // lds_addr (bytes)
        g0[2] = (uint32_t)gaddr;                      // global_addr[31:0]
        g0[3] = (uint32_t)((gaddr >> 32) & 0x01FFFFFFu) | (2u << 30);  // type=2

        i32x8 g1;
        g1[0] = (int)(2u << 16);                      // data_size = 4B; mask=0
        g1[1] = (int)((tdim0 & 0xFFFFu) << 16);       // tensor_dim0[15:0] @63:48
        g1[2] = (int)(((tdim0 >> 16) & 0xFFFFu) | ((nLine & 0xFFFFu) << 16));
        g1[3] = (int)(((nLine >> 16) & 0xFFFFu) | ((tdim0 & 0xFFFFu) << 16));
        g1[4] = (int)(nLine & 0xFFFFu);               // tile_dim1 (tile_dim2 = 0)
        g1[5] = (int)lstr;                            // tensor_dim0_stride[31:0]
        g1[6] = 0;
        g1[7] = 0;

        i32x4 gz = {0, 0, 0, 0};
#if __clang_major__ >= 23
        i32x8 gp = {0, 0, 0, 0, 0, 0, 0, 0};
        __builtin_amdgcn_tensor_load_to_lds(g0, g1, gz, gz, gp, 0);
#else
        __builtin_amdgcn_tensor_load_to_lds(g0, g1, gz, gz, 0);
#endif
    }
#endif

    const int kb  = (lane >> 4) << 1;     // K sub-base: 0 or 2
    const int col = lane & 15;

    // A fragments: 12 coalesced b64 loads from the per-lane table
    v2f A[3][4];
#pragma unroll
    for (int o = 0; o < 3; ++o)
#pragma unroll
        for (int q = 0; q < 4; ++q)
            A[o][q] = *(const v2f*)(aw + ((size_t)((o * 4 + q) * 32 + lane)) * 2);

#if HAVE_TDM
    __builtin_amdgcn_s_wait_tensorcnt(0);
    asm volatile("" ::: "memory");        // keep LDS reads after the wait
    const float* lbase = &tile[wave][0];
#else
    const float* gbase = in + elemOff;
#endif

    v8f acc = {0.f, 0.f, 0.f, 0.f, 0.f, 0.f, 0.f, 0.f};

    // Center tile (always valid), then band neighbors under scalar guards.
#if HAVE_TDM
    acc = tile_mac<CONTIG>(acc, A[1], lbase, nAxis, (axisTile - jStart) * 16, kb, col);
    if (axisTile > 0)
        acc = tile_mac<CONTIG>(acc, A[0], lbase, nAxis, 0, kb, col);
    if (axisTile < 5)
        acc = tile_mac<CONTIG>(acc, A[2], lbase, nAxis, (axisTile + 1 - jStart) * 16, kb, col);
#else
    acc = tile_mac<AS, CS>(acc, A[1], gbase, (axisTile - jStart) * 16, kb, col);
    if (axisTile > 0)
        acc = tile_mac<AS, CS>(acc, A[0], gbase, 0, kb, col);
    if (axisTile < 5)
        acc = tile_mac<AS, CS>(acc, A[2], gbase, (axisTile + 1 - jStart) * 16, kb, col);
#endif

    // D store: VGPR r = row r (lanes 0-15) / r+8 (lanes 16-31), col = lane&15
    const int rAdd = (lane >> 4) << 3;
    float* op = out + (size_t)ch * NVOX + (size_t)plane * PS
                    + (size_t)colTile * 16 * CS
                    + (size_t)axisTile * 16 * AS
                    + (size_t)col * CS;
    if (CONTIG) {   // rows contiguous along axis -> two b128 stores
        float4 lo = {acc[0], acc[1], acc[2], acc[3]};
        float4 hi = {acc[4], acc[5], acc[6], acc[7]};
        *(float4*)(op + rAdd) = lo;
        *(float4*)(op + rAdd + 4) = hi;
    } else {
#pragma unroll
        for (int r = 0; r < 8; ++r)
            op[(size_t)(r + rAdd) * AS] = acc[r];
    }
}

// ---------------------------------------------------------------------------
// Central-difference gradient (replicate padding); also emit rg = -res * g.
// ---------------------------------------------------------------------------
__global__ __launch_bounds__(256) void grad_kernel(
    const float* __restrict__ img, const float* __restrict__ res,
    float* __restrict__ gOut, float* __restrict__ rg) {
    int idx = blockIdx.x * 256 + threadIdx.x;
    if (idx >= NVOX) return;
    int x = idx % S1;
    int y = (idx / S1) % DIMV;
    int z = idx / S2;
    int xm = max(x - 1, 0), xp = min(x + 1, DIMV - 1);
    int ym = max(y - 1, 0), yp = min(y + 1, DIMV - 1);
    int zm = max(z - 1, 0), zp = min(z + 1, DIMV - 1);
    int rowBase = z * S2 + y * S1;
    float dx = 0.5f * (img[rowBase + xp] - img[rowBase + xm]);
    float dy = 0.5f * (img[z * S2 + yp * S1 + x] - img[z * S2 + ym * S1 + x]);
    float dz = 0.5f * (img[zp * S2 + y * S1 + x] - img[zm * S2 + y * S1 + x]);
    float r = res[idx];
    gOut[idx] = dx;
    gOut[NVOX + idx] = dy;
    gOut[2 * (size_t)NVOX + idx] = dz;
    rg[idx] = -r * dx;
    rg[NVOX + idx] = -r * dy;
    rg[2 * (size_t)NVOX + idx] = -r * dz;
}

// fields output: v_p[b,x,y,z,c] = v[c,z,y,x]
__global__ __launch_bounds__(256) void fields_kernel(
    const float* __restrict__ v, float* __restrict__ f) {
    int idx = blockIdx.x * 256 + threadIdx.x;
    if (idx >= NVOX) return;
    int x = idx % S1;
    int y = (idx / S1) % DIMV;
    int z = idx / S2;
    size_t o = ((size_t)(x * S1 + y) * S1 + z) * 3;
    f[o + 0] = v[idx];
    f[o + 1] = v[NVOX + idx];
    f[o + 2] = v[2 * (size_t)NVOX + idx];
}

// f = div(res*v)/L (3x3x3 Sobel /32, zero padding); res_next = res - f
__global__ __launch_bounds__(256) void div_kernel(
    const float* __restrict__ res, const float* __restrict__ v,
    float* __restrict__ resNext) {
    int idx = blockIdx.x * 256 + threadIdx.x;
    if (idx >= NVOX) return;
    int x = idx % S1;
    int y = (idx / S1) % DIMV;
    int z = idx / S2;
    const float sW[3] = {1.f, 2.f, 1.f};
    const float dW[3] = {-1.f, 0.f, 1.f};
    float acc = 0.0f;
#pragma unroll
    for (int dz = 0; dz < 3; ++dz) {
        int zz = z + dz - 1;
        if (zz < 0 || zz >= DIMV) continue;
#pragma unroll
        for (int dy = 0; dy < 3; ++dy) {
            int yy = y + dy - 1;
            if (yy < 0 || yy >= DIMV) continue;
#pragma unroll
            for (int dx = 0; dx < 3; ++dx) {
                int xx = x + dx - 1;
                if (xx < 0 || xx >= DIMV) continue;
                int n = zz * S2 + yy * S1 + xx;
                float rr = res[n];
                float wx = sW[dz] * sW[dy] * dW[dx];
                float wy = sW[dz] * dW[dy] * sW[dx];
                float wz = dW[dz] * sW[dy] * sW[dx];
                acc += rr * (wx * v[n] + wy * v[NVOX + n]
                           + wz * v[2 * (size_t)NVOX + n]);
            }
        }
    }
    resNext[idx] = res[idx] - acc * (1.0f / 32.0f) * (1.0f / (float)LSTEPS);
}

// Trilinear sample helper (reference clip semantics: frac before index clip)
struct Tri {
    int i000, i001, i010, i011, i100, i101, i110, i111;
    float w000, w001, w010, w011, w100, w101, w110, w111;
};

__device__ __forceinline__ Tri make_tri(float x, float y, float z) {
    float x0f = floorf(x), y0f = floorf(y), z0f = floorf(z);
    float fx = x - x0f, fy = y - y0f, fz = z - z0f;
    int x0 = min(max((int)x0f, 0), DIMV - 1);
    int y0 = min(max((int)y0f, 0), DIMV - 1);
    int z0 = min(max((int)z0f, 0), DIMV - 1);
    int x1 = min(x0 + 1, DIMV - 1);
    int y1 = min(y0 + 1, DIMV - 1);
    int z1 = min(z0 + 1, DIMV - 1);
    Tri s;
    s.i000 = z0 * S2 + y0 * S1 + x0; s.i001 = z0 * S2 + y0 * S1 + x1;
    s.i010 = z0 * S2 + y1 * S1 + x0; s.i011 = z0 * S2 + y1 * S1 + x1;
    s.i100 = z1 * S2 + y0 * S1 + x0; s.i101 = z1 * S2 + y0 * S1 + x1;
    s.i110 = z1 * S2 + y1 * S1 + x0; s.i111 = z1 * S2 + y1 * S1 + x1;
    float gx = 1.f - fx, gy = 1.f - fy, gz = 1.f - fz;
    s.w000 = gx * gy * gz; s.w001 = fx * gy * gz;
    s.w010 = gx * fy * gz; s.w011 = fx * fy * gz;
    s.w100 = gx * gy * fz; s.w101 = fx * gy * fz;
    s.w110 = gx * fy * fz; s.w111 = fx * fy * fz;
    return s;
}

__device__ __forceinline__ float tri_apply(const Tri& s, const float* __restrict__ p) {
    return p[s.i000] * s.w000 + p[s.i001] * s.w001 + p[s.i010] * s.w010 +
           p[s.i011] * s.w011 + p[s.i100] * s.w100 + p[s.i101] * s.w101 +
           p[s.i110] * s.w110 + p[s.i111] * s.w111;
}

// Fused: resdef_new = deform(resdef, id - v/L) + res_next; phi_new = deform(phi)
__global__ __launch_bounds__(256) void deform_step(
    const float* __restrict__ v, const float* __restrict__ resNext,
    const float* __restrict__ rdIn, const float* __restrict__ phIn,
    float* __restrict__ rdOut, float* __restrict__ phOut) {
    int idx = blockIdx.x * 256 + threadIdx.x;
    if (idx >= NVOX) return;
    int a2 = idx % S1;
    int a1 = (idx / S1) % DIMV;
    int a0 = idx / S2;
    int tIdx = a2 * S2 + a1 * S1 + a0;            // transposed (cubic volume)
    const float s = (float)(DIMV - 1) / (2.0f * (float)LSTEPS);   // 4.75
    float x = (float)a2 - v[tIdx] * s;
    float y = (float)a1 - v[NVOX + tIdx] * s;
    float z = (float)a0 - v[2 * (size_t)NVOX + tIdx] * s;
    Tri tr = make_tri(x, y, z);
    rdOut[idx] = tri_apply(tr, rdIn) + resNext[idx];
    phOut[idx] = tri_apply(tr, phIn);
    phOut[NVOX + idx] = tri_apply(tr, phIn + NVOX);
    phOut[2 * (size_t)NVOX + idx] = tri_apply(tr, phIn + 2 * (size_t)NVOX);
}

// Fused: image = deform(source, phi) + resdef * (MU^2/L) * deform(seg, phi)
__global__ __launch_bounds__(256) void image_update(
    const float* __restrict__ src, const float* __restrict__ seg,
    const float* __restrict__ phi, const float* __restrict__ rdef,
    float* __restrict__ img) {
    int idx = blockIdx.x * 256 + threadIdx.x;
    if (idx >= NVOX) return;
    int a2 = idx % S1;
    int a1 = (idx / S1) % DIMV;
    int a0 = idx / S2;
    int tIdx = a2 * S2 + a1 * S1 + a0;
    const float h = 0.5f * (float)(DIMV - 1);
    float x = (phi[tIdx] + 1.0f) * h;
    float y = (phi[NVOX + tIdx] + 1.0f) * h;
    float z = (phi[2 * (size_t)NVOX + tIdx] + 1.0f) * h;
    Tri tr = make_tri(x, y, z);
    float sv = tri_apply(tr, src);
    float mv = tri_apply(tr, seg);
    img[idx] = sv + rdef[idx] * 0.001f * mv;      // MU^2 / L = 0.01 / 10
}

// phi init (planar transposed layout)
__global__ __launch_bounds__(256) void phi_init(float* __restrict__ phi) {
    int idx = blockIdx.x * 256 + threadIdx.x;
    if (idx >= NVOX) return;
    int r = idx % S1;
    int q = (idx / S1) % DIMV;
    int p = idx / S2;
    const float sc = 2.0f / (float)(DIMV - 1);
    phi[idx] = -1.0f + sc * (float)p;
    phi[NVOX + idx] = -1.0f + sc * (float)q;
    phi[2 * (size_t)NVOX + idx] = -1.0f + sc * (float)r;
}

extern "C" void kernel_launch(void* const* d_in, const int* in_sizes, int n_in,
                              void* d_out, int out_size, void* d_ws, size_t ws_size,
                              hipStream_t stream) {
    const float* src = (const float*)d_in[0];
    // d_in[1] (target) is unused by the reference computation
    const float* seg = (const float*)d_in[2];
    const float* z0  = (const float*)d_in[3];
    float* out = (float*)d_out;
    float* ws  = (float*)d_ws;

    const size_t N = (size_t)NVOX;
    float* rgTmp    = ws;                 // 3N
    float* tmpB     = ws + 3 * N;         // 3N (final v lives here)
    float* imageBuf = ws + 6 * N;         // N
    float* rdA      = ws + 7 * N;         // N
    float* rdB      = ws + 8 * N;         // N
    float* phA      = ws + 9 * N;         // 3N
    float* phB      = ws + 12 * N;        // 3N
    float* wsA      = ws + 15 * N;        // 768 floats: WMMA A-fragment table

    float* outImage  = out;
    float* outFields = out + N;
    float* outGrads  = outFields + (size_t)LSTEPS * 3 * N;
    float* outResid  = outGrads + (size_t)LSTEPS * 3 * N;
    float* outResdef = outResid + (size_t)(LSTEPS + 1) * N;

    const int EB = 256, EG = NVOX / 256;              // 3456 blocks, exact
    const int BB = 128, BG = (3 * DIMV * 6 * 6) / 4;  // 10368 waves / 4 = 2592

    hipMemcpyAsync(imageBuf, src, N * sizeof(float), hipMemcpyDeviceToDevice, stream);
    hipMemcpyAsync(outResid, z0, N * sizeof(float), hipMemcpyDeviceToDevice, stream);
    hipMemsetAsync(rdA, 0, N * sizeof(float), stream);
    phi_init<<<EG, EB, 0, stream>>>(phA);
    ainit_kernel<<<1, 384, 0, stream>>>(wsA);

    float* rdCur = rdA; float* rdNxt = rdB;
    float* phCur = phA; float* phNxt = phB;

    for (int i = 0; i < LSTEPS; ++i) {
        const float* res = outResid + (size_t)i * N;
        float* resNext   = outResid + (size_t)(i + 1) * N;

        grad_kernel<<<EG, EB, 0, stream>>>(imageBuf, res,
                                           outGrads + (size_t)i * 3 * N, rgTmp);

        // separable Gaussian via fp32 WMMA + TDM staging: W, H, D axes
        blur_axis_wmma<1, S1, S2, true><<<BG, BB, 0, stream>>>(rgTmp, tmpB, wsA);
        blur_axis_wmma<S1, 1, S2, false><<<BG, BB, 0, stream>>>(tmpB, rgTmp, wsA);
        blur_axis_wmma<S2, 1, S1, false><<<BG, BB, 0, stream>>>(rgTmp, tmpB, wsA);
        float* v = tmpB;

        fields_kernel<<<EG, EB, 0, stream>>>(v, outFields + (size_t)i * 3 * N);
        div_kernel<<<EG, EB, 0, stream>>>(res, v, resNext);
        deform_step<<<EG, EB, 0, stream>>>(v, resNext, rdCur, phCur, rdNxt, phNxt);
        { float* t = rdCur; rdCur = rdNxt; rdNxt = t; }
        { float* t = phCur; phCur = phNxt; phNxt = t; }
        image_update<<<EG, EB, 0, stream>>>(src, seg, phCur, rdCur, imageBuf);
    }

    hipMemcpyAsync(outImage, imageBuf, N * sizeof(float), hipMemcpyDeviceToDevice, stream);
    hipMemcpyAsync(outResdef, rdCur, N * sizeof(float), hipMemcpyDeviceToDevice, stream);
}